// GAT_V2_87144886436015
// MI455X (gfx1250) — compile-verified
//
#include <hip/hip_runtime.h>
#include <math.h>

typedef float v2f __attribute__((ext_vector_type(2)));
typedef float v8f __attribute__((ext_vector_type(8)));

#define GAT_N 50000
#define GAT_E 800000
#define GAT_H 4
#define GAT_C 32
#define GAT_HC 128
#define GAT_ED 16
#define NEG_SLOPE 0.2f

// ---------------- utility ----------------
__global__ void fill_f32_kernel(float* __restrict__ p, size_t n, float v) {
  size_t t = (size_t)blockIdx.x * blockDim.x + threadIdx.x;
  if (t < n) p[t] = v;
}

// accumulate per-dst edge_attr sums + counts (for self-loop 'mean' fill)
__global__ void loop_accum_kernel(const int* __restrict__ ei, const float* __restrict__ eattr,
                                  float* __restrict__ cnt, float* __restrict__ sums) {
  size_t t = (size_t)blockIdx.x * blockDim.x + threadIdx.x;
  if (t >= (size_t)GAT_E * GAT_ED) return;
  int e = (int)(t >> 4), k = (int)(t & 15);
  int dst = ei[GAT_E + e];
  atomicAdd(&sums[(size_t)dst * GAT_ED + k], eattr[t]);
  if (k == 0) atomicAdd(&cnt[dst], 1.0f);
}

__global__ void loop_finish_kernel(const float* __restrict__ cnt, const float* __restrict__ sums,
                                   float* __restrict__ lattr) {
  size_t t = (size_t)blockIdx.x * blockDim.x + threadIdx.x;
  if (t >= (size_t)GAT_N * GAT_ED) return;
  int n = (int)(t >> 4);
  lattr[t] = sums[t] / fmaxf(cnt[n], 1.0f);
}

// ---------------- fp32 WMMA GEMM: out[M,Nout] = A[M,K] @ W[K,Nout] + bias ----------------
// One wave per 16x16 output tile. A-tile 16x4 f32: lanes 0-15 rows M, VGPR0/1 = K{0,1};
// lanes 16-31 = K{2,3}. B-tile 4x16 mirrored. D: VGPR r -> row (r | +8 for upper lanes).
__global__ void wmma_linear_kernel(const float* __restrict__ A, const float* __restrict__ W,
                                   const float* __restrict__ bias, float* __restrict__ out,
                                   int M, int K, int Nout) {
  int lane = threadIdx.x & 31;
  int wave = (int)(((size_t)blockIdx.x * blockDim.x + threadIdx.x) >> 5);
  int tiles_n = Nout >> 4;
  int tiles = (M >> 4) * tiles_n;
  if (wave >= tiles) return;                 // wave-uniform: EXEC stays all-1s for WMMA
  int tm = wave / tiles_n, tn = wave - tm * tiles_n;
  int rowA = tm * 16 + (lane & 15);
  int koff = (lane >> 4) << 1;               // 0 or 2
  int col  = tn * 16 + (lane & 15);
  const float* arow = A + (size_t)rowA * K;
  v8f acc = {};
  for (int k = 0; k < K; k += 4) {
    v2f a = *(const v2f*)(arow + k + koff);  // contiguous K-pair, 8B aligned
    v2f b;
    b.x = W[(size_t)(k + koff) * Nout + col];
    b.y = W[(size_t)(k + koff + 1) * Nout + col];
    acc = __builtin_amdgcn_wmma_f32_16x16x4_f32(false, a, false, b, (short)0, acc,
                                                false, false);
  }
  float bv = bias[col];
  int mbase = tm * 16 + ((lane >> 4) << 3);
  #pragma unroll
  for (int r = 0; r < 8; ++r)
    out[(size_t)(mbase + r) * Nout + col] = acc[r] + bv;
}

// ---------------- edge passes (wave32: channel = head*32 + lane) ----------------
__device__ __forceinline__ void atomicMaxF(float* addr, float val) {
  int iv = __float_as_int(val);
  if (iv >= 0) atomicMax((int*)addr, iv);
  else         atomicMin((unsigned int*)addr, (unsigned int)iv);
}

// pass1: s[e,h] = sum_c att[h,c] * leakyrelu(xl[src]+xr[dst]+eattr@We); atomic segment max
__global__ void edge_pass1_kernel(const int* __restrict__ ei, const float* __restrict__ eattr,
                                  const float* __restrict__ lattr,
                                  const float* __restrict__ xl, const float* __restrict__ xr,
                                  const float* __restrict__ We, const float* __restrict__ att,
                                  float* __restrict__ sbuf, float* __restrict__ smax) {
  size_t g = (size_t)blockIdx.x * blockDim.x + threadIdx.x;
  int wid = (int)(g >> 5);
  int lane = threadIdx.x & 31;
  const int Etot = GAT_E + GAT_N;
  if (wid >= Etot) return;
  int src, dst;
  const float* ea;
  if (wid < GAT_E) {
    src = ei[wid]; dst = ei[GAT_E + wid];
    ea = eattr + (size_t)wid * GAT_ED;
  } else {
    src = dst = wid - GAT_E;
    ea = lattr + (size_t)(wid - GAT_E) * GAT_ED;
  }
  float eav = (lane < GAT_ED) ? ea[lane] : 0.0f;   // lanes 0..15 hold edge features
  const float* xls = xl + (size_t)src * GAT_HC;
  const float* xrd = xr + (size_t)dst * GAT_HC;
  float s[GAT_H];
  #pragma unroll
  for (int h = 0; h < GAT_H; ++h) {
    int c = h * GAT_C + lane;
    float ev = 0.0f;
    #pragma unroll
    for (int k = 0; k < GAT_ED; ++k)
      ev = fmaf(__shfl(eav, k, 32), We[k * GAT_HC + c], ev);  // in-register eattr@We
    float m = xls[c] + xrd[c] + ev;
    m = (m > 0.0f) ? m : NEG_SLOPE * m;
    s[h] = att[c] * m;                               // att flat index == channel
  }
  #pragma unroll
  for (int off = 16; off > 0; off >>= 1) {
    #pragma unroll
    for (int h = 0; h < GAT_H; ++h) s[h] += __shfl_xor(s[h], off, 32);
  }
  if (lane == 0) {
    #pragma unroll
    for (int h = 0; h < GAT_H; ++h) {
      sbuf[(size_t)wid * GAT_H + h] = s[h];
      atomicMaxF(&smax[(size_t)dst * GAT_H + h], s[h]);
    }
  }
}

// pass2: ex = exp(s - smax[dst]); denom[dst] += ex
__global__ void edge_pass2_kernel(const int* __restrict__ ei, const float* __restrict__ smax,
                                  float* __restrict__ sbuf, float* __restrict__ denom) {
  size_t t = (size_t)blockIdx.x * blockDim.x + threadIdx.x;
  const size_t tot = (size_t)(GAT_E + GAT_N) * GAT_H;
  if (t >= tot) return;
  int e = (int)(t >> 2), h = (int)(t & 3);
  int dst = (e < GAT_E) ? ei[GAT_E + e] : (e - GAT_E);
  float ex = expf(sbuf[t] - smax[(size_t)dst * GAT_H + h]);
  sbuf[t] = ex;
  atomicAdd(&denom[(size_t)dst * GAT_H + h], ex);
}

// pass3: agg[dst] += (ex/denom[dst]) * xl[src]
__global__ void edge_pass3_kernel(const int* __restrict__ ei, const float* __restrict__ sbuf,
                                  const float* __restrict__ denom, const float* __restrict__ xl,
                                  float* __restrict__ agg) {
  size_t g = (size_t)blockIdx.x * blockDim.x + threadIdx.x;
  int wid = (int)(g >> 5);
  int lane = threadIdx.x & 31;
  const int Etot = GAT_E + GAT_N;
  if (wid >= Etot) return;
  int src, dst;
  if (wid < GAT_E) { src = ei[wid]; dst = ei[GAT_E + wid]; }
  else             { src = dst = wid - GAT_E; }
  const float* xls = xl + (size_t)src * GAT_HC;
  float* ag = agg + (size_t)dst * GAT_HC;
  #pragma unroll
  for (int h = 0; h < GAT_H; ++h) {
    float alpha = sbuf[(size_t)wid * GAT_H + h] / denom[(size_t)dst * GAT_H + h];
    atomicAdd(&ag[h * GAT_C + lane], alpha * xls[h * GAT_C + lane]);
  }
}

__global__ void finalize_kernel(const float* __restrict__ agg, const float* __restrict__ bias,
                                float* __restrict__ out, int do_elu) {
  size_t t = (size_t)blockIdx.x * blockDim.x + threadIdx.x;
  if (t >= (size_t)GAT_N * GAT_HC) return;
  int c = (int)(t & (GAT_HC - 1));
  float v = agg[t] + bias[c];
  if (do_elu) v = (v > 0.0f) ? v : (expf(v) - 1.0f);
  out[t] = v;
}

// ---------------- host ----------------
static inline unsigned gdim(size_t total, unsigned block) {
  return (unsigned)((total + block - 1) / block);
}

extern "C" void kernel_launch(void* const* d_in, const int* in_sizes, int n_in,
                              void* d_out, int out_size, void* d_ws, size_t ws_size,
                              hipStream_t stream) {
  const float* x     = (const float*)d_in[0];
  const int*   ei    = (const int*)d_in[1];
  const float* eattr = (const float*)d_in[2];
  const float* Wl1 = (const float*)d_in[3];  const float* bl1 = (const float*)d_in[4];
  const float* Wr1 = (const float*)d_in[5];  const float* br1 = (const float*)d_in[6];
  const float* We1 = (const float*)d_in[7];  const float* att1 = (const float*)d_in[8];
  const float* bias1 = (const float*)d_in[9];
  const float* Wl2 = (const float*)d_in[10]; const float* bl2 = (const float*)d_in[11];
  const float* Wr2 = (const float*)d_in[12]; const float* br2 = (const float*)d_in[13];
  const float* We2 = (const float*)d_in[14]; const float* att2 = (const float*)d_in[15];
  const float* bias2 = (const float*)d_in[16];
  const float* Wlin = (const float*)d_in[17]; const float* blin = (const float*)d_in[18];
  float* out = (float*)d_out;
  (void)in_sizes; (void)n_in; (void)out_size; (void)ws_size;

  // workspace carve-out (~150 MB total; all arrays L2-resident on 192MB L2)
  char* w = (char*)d_ws;
  auto carve = [&](size_t bytes) -> float* {
    float* p = (float*)w; w += (bytes + 255) & ~(size_t)255; return p;
  };
  float* cnt   = carve((size_t)GAT_N * 4);
  float* sums  = carve((size_t)GAT_N * GAT_ED * 4);
  float* lattr = carve((size_t)GAT_N * GAT_ED * 4);
  float* xl    = carve((size_t)GAT_N * GAT_HC * 4);
  float* xr    = carve((size_t)GAT_N * GAT_HC * 4);
  float* sbuf  = carve((size_t)(GAT_E + GAT_N) * GAT_H * 4);
  float* smax  = carve((size_t)GAT_N * GAT_H * 4);
  float* denom = carve((size_t)GAT_N * GAT_H * 4);
  float* agg   = carve((size_t)GAT_N * GAT_HC * 4);
  float* h1    = carve((size_t)GAT_N * GAT_HC * 4);
  float* h2    = carve((size_t)GAT_N * GAT_HC * 4);

  const unsigned B = 256;
  const size_t Etot = (size_t)GAT_E + GAT_N;

  // self-loop edge_attr = mean of incoming edge_attr
  fill_f32_kernel<<<gdim((size_t)GAT_N, B), B, 0, stream>>>(cnt, (size_t)GAT_N, 0.0f);
  fill_f32_kernel<<<gdim((size_t)GAT_N * GAT_ED, B), B, 0, stream>>>(sums, (size_t)GAT_N * GAT_ED, 0.0f);
  loop_accum_kernel<<<gdim((size_t)GAT_E * GAT_ED, B), B, 0, stream>>>(ei, eattr, cnt, sums);
  loop_finish_kernel<<<gdim((size_t)GAT_N * GAT_ED, B), B, 0, stream>>>(cnt, sums, lattr);

  auto run_gemm = [&](const float* A, const float* W_, const float* b_, float* o,
                      int M, int K, int Nout) {
    size_t waves = (size_t)(M >> 4) * (Nout >> 4);
    wmma_linear_kernel<<<gdim(waves * 32, B), B, 0, stream>>>(A, W_, b_, o, M, K, Nout);
  };
  auto run_layer = [&](const float* Xin, const float* Wl, const float* bl,
                       const float* Wr, const float* br, const float* We,
                       const float* att, const float* bias, float* hout, int do_elu) {
    run_gemm(Xin, Wl, bl, xl, GAT_N, GAT_HC, GAT_HC);
    run_gemm(Xin, Wr, br, xr, GAT_N, GAT_HC, GAT_HC);
    fill_f32_kernel<<<gdim((size_t)GAT_N * GAT_H, B), B, 0, stream>>>(smax, (size_t)GAT_N * GAT_H, -INFINITY);
    fill_f32_kernel<<<gdim((size_t)GAT_N * GAT_H, B), B, 0, stream>>>(denom, (size_t)GAT_N * GAT_H, 0.0f);
    fill_f32_kernel<<<gdim((size_t)GAT_N * GAT_HC, B), B, 0, stream>>>(agg, (size_t)GAT_N * GAT_HC, 0.0f);
    edge_pass1_kernel<<<gdim(Etot * 32, B), B, 0, stream>>>(ei, eattr, lattr, xl, xr, We, att, sbuf, smax);
    edge_pass2_kernel<<<gdim(Etot * GAT_H, B), B, 0, stream>>>(ei, smax, sbuf, denom);
    edge_pass3_kernel<<<gdim(Etot * 32, B), B, 0, stream>>>(ei, sbuf, denom, xl, agg);
    finalize_kernel<<<gdim((size_t)GAT_N * GAT_HC, B), B, 0, stream>>>(agg, bias, hout, do_elu);
  };

  run_layer(x,  Wl1, bl1, Wr1, br1, We1, att1, bias1, h1, /*elu=*/1);
  run_layer(h1, Wl2, bl2, Wr2, br2, We2, att2, bias2, h2, /*elu=*/0);
  run_gemm(h2, Wlin, blin, out, GAT_N, GAT_HC, 16);   // final linear -> d_out [N,16]
}